// Initial_h_48215302865401
// MI455X (gfx1250) — compile-verified
//
#include <hip/hip_runtime.h>
#include <hip/hip_bf16.h>
#include <stdint.h>

// ---- problem constants (match reference) ----
#define NUM_ENTS   80000
#define NUM_WORDS  20000
#define NNODES     100000
#define HDIM       200
#define NBLK       100            // 2x2 blocks per row
#define NRELS      16
#define NEDGES     300000
#define RRELU_SLOPE ((1.0f/8.0f + 1.0f/3.0f) * 0.5f)

#define WELEMS     (NRELS * NBLK * 4)   // 6400 floats = 25600 bytes

typedef unsigned int v4u __attribute__((ext_vector_type(4)));
typedef int          v8i __attribute__((ext_vector_type(8)));
typedef int          v4i __attribute__((ext_vector_type(4)));

// Hardware f32 atomic add (global_atomic_add_f32), avoid CAS loop.
__device__ __forceinline__ void atomicAddF32(float* p, float v) {
#if defined(__HIP_DEVICE_COMPILE__)
    unsafeAtomicAdd(p, v);
#else
    atomicAdd(p, v);
#endif
}

// ---------------- kernel 1: zero agg + indeg ----------------
__global__ __launch_bounds__(256)
void zero_ws_kernel(float4* __restrict__ ws, int n4) {
    int i      = blockIdx.x * blockDim.x + threadIdx.x;
    int stride = gridDim.x * blockDim.x;
    float4 z = make_float4(0.f, 0.f, 0.f, 0.f);
    for (; i < n4; i += stride) ws[i] = z;
}

// ---------------- kernel 2: edge gather / 2x2 block matmul / scatter-add ----------------
__device__ __forceinline__ void edge_block(const float2* __restrict__ hrow,
                                           const float4* __restrict__ wrow,
                                           float* __restrict__ arow, int b) {
    float2 hv = hrow[b];           // h[2b], h[2b+1]              (global_load_b64, L2-resident)
    float4 wv = wrow[b];           // (w00, w01, w10, w11)        (ds_load_b128)
    float m0 = fmaf(hv.y, wv.z, hv.x * wv.x);   // o=0
    float m1 = fmaf(hv.y, wv.w, hv.x * wv.y);   // o=1
    atomicAddF32(arow + 2 * b,     m0);         // global_atomic_add_f32 -> L2 atomic unit
    atomicAddF32(arow + 2 * b + 1, m1);
}

__global__ __launch_bounds__(256)
void edge_kernel(const float* __restrict__ dyn,
                 const float* __restrict__ wrd,
                 const float* __restrict__ weight,
                 const int*   __restrict__ src,
                 const int*   __restrict__ dst,
                 const int*   __restrict__ ety,
                 float*       __restrict__ agg,
                 float*       __restrict__ indeg)
{
    // 25.6 KB weight table staged in LDS (sole LDS object -> LDS byte offset 0)
    __shared__ float4 w_lds[NRELS * NBLK];

    // ---- TDM: one descriptor-driven DMA per workgroup (wave 0 issues it once) ----
    if (threadIdx.x < 32) {
        uint64_t ga = (uint64_t)(uintptr_t)weight;
        // D# group0: count=1, lds_addr=0, global_addr, type=2 ("image")
        v4u g0 = { 1u,
                   0u,
                   (unsigned)(ga & 0xFFFFFFFFu),
                   (unsigned)((ga >> 32) & 0x01FFFFFFu) | (2u << 30) };
        // D# group1: data_size=4B (code 2); 1-D tile: tensor_dim0 = tile_dim0 = 6400,
        // tensor_dim1 = tile_dim1 = 1, tensor_dim0_stride = 6400
        v8i g1 = { (int)(2u << 16),              // workgroup_mask=0, data_size=2 (4B)
                   (int)((unsigned)WELEMS << 16),// tensor_dim0[15:0] in bits 63:48
                   (int)(1u << 16),              // tensor_dim0[31:16]=0, tensor_dim1=1
                   (int)((unsigned)WELEMS << 16),// tensor_dim1[31:16]=0, tile_dim0=6400
                   1,                            // tile_dim1=1, tile_dim2=0
                   WELEMS,                       // tensor_dim0_stride[31:0]
                   0, 0 };
        v4i zg4 = { 0, 0, 0, 0 };
        v8i zg8 = { 0, 0, 0, 0, 0, 0, 0, 0 };
        __builtin_amdgcn_tensor_load_to_lds(g0, g1, zg4, zg4, zg8, 0); // -> tensor_load_to_lds
        __builtin_amdgcn_s_wait_tensorcnt(0);                          // -> s_wait_tensorcnt
    }
    __syncthreads();

    const int lane   = threadIdx.x & 31;
    const int wid    = threadIdx.x >> 5;
    int       gw     = blockIdx.x * (blockDim.x >> 5) + wid;
    const int nwaves = gridDim.x * (blockDim.x >> 5);

    for (int e0 = gw; e0 < NEDGES; e0 += nwaves) {
        int e = __builtin_amdgcn_readfirstlane(e0);   // wave-uniform -> scalar loads
        int s = src[e];
        int d = dst[e];
        int t = ety[e];

        const float* hbase = (s < NUM_ENTS) ? (dyn + (size_t)s * HDIM)
                                            : (wrd + (size_t)(s - NUM_ENTS) * HDIM);
        const float2* hrow = (const float2*)hbase;
        const float4* wrow = w_lds + t * NBLK;
        float*        arow = agg + (size_t)d * HDIM;

        // 100 blocks over 32 lanes: 3 full sweeps + 4-lane tail
        #pragma unroll
        for (int k = 0; k < 3; ++k)
            edge_block(hrow, wrow, arow, lane + 32 * k);
        if (lane < 4)
            edge_block(hrow, wrow, arow, lane + 96);

        if (lane == 0)
            atomicAddF32(indeg + d, 1.0f);
    }
}

// ---------------- kernel 3: mean-normalize, leaky ReLU, L2 row-normalize, duplicate ----------------
__device__ __forceinline__ float2 rrelu2(float2 v, float scl) {
    float a = v.x * scl;
    float b = v.y * scl;
    a = (a >= 0.f) ? a : a * RRELU_SLOPE;
    b = (b >= 0.f) ? b : b * RRELU_SLOPE;
    return make_float2(a, b);
}

__global__ __launch_bounds__(256)
void finalize_kernel(const float* __restrict__ agg,
                     const float* __restrict__ indeg,
                     float*       __restrict__ out)
{
    const int lane = threadIdx.x & 31;
    const int row  = blockIdx.x * 8 + (threadIdx.x >> 5);
    if (row >= NUM_ENTS) return;

    float deg = indeg[row];
    float scl = (deg > 0.f) ? (1.0f / fmaxf(deg, 1.0f)) : 0.0f;

    // row = 100 float2 elements: 3 full 32-lane sweeps + 4-lane tail
    const float2* arow = (const float2*)(agg + (size_t)row * HDIM);
    float2 vals[4];
    float  ss = 0.f;
    #pragma unroll
    for (int k = 0; k < 3; ++k) {
        float2 v = rrelu2(arow[lane + 32 * k], scl);
        vals[k] = v;
        ss = fmaf(v.y, v.y, fmaf(v.x, v.x, ss));
    }
    if (lane < 4) {
        float2 v = rrelu2(arow[96 + lane], scl);
        vals[3] = v;
        ss = fmaf(v.y, v.y, fmaf(v.x, v.x, ss));
    }

    // wave32 reduction of sum of squares
    #pragma unroll
    for (int off = 16; off > 0; off >>= 1)
        ss += __shfl_xor(ss, off, 32);

    float inv = 1.0f / fmaxf(sqrtf(ss), 1e-12f);

    float2* o1 = (float2*)(out + (size_t)row * HDIM);
    float2* o2 = (float2*)(out + (size_t)NUM_ENTS * HDIM + (size_t)row * HDIM);
    #pragma unroll
    for (int k = 0; k < 3; ++k) {
        float2 v = make_float2(vals[k].x * inv, vals[k].y * inv);
        o1[lane + 32 * k] = v;
        o2[lane + 32 * k] = v;
    }
    if (lane < 4) {
        float2 v = make_float2(vals[3].x * inv, vals[3].y * inv);
        o1[96 + lane] = v;
        o2[96 + lane] = v;
    }
}

// ---------------- host launcher ----------------
extern "C" void kernel_launch(void* const* d_in, const int* in_sizes, int n_in,
                              void* d_out, int out_size, void* d_ws, size_t ws_size,
                              hipStream_t stream) {
    const float* dyn    = (const float*)d_in[0];   // (80000, 200) f32
    const float* wrd    = (const float*)d_in[1];   // (20000, 200) f32
    const float* weight = (const float*)d_in[2];   // (16, 100, 2, 2) f32
    const int*   src    = (const int*)d_in[3];     // (300000,) i32
    const int*   dst    = (const int*)d_in[4];     // (300000,) i32
    const int*   ety    = (const int*)d_in[5];     // (300000,) i32
    float*       out    = (float*)d_out;           // 2 * 80000 * 200 f32

    // workspace layout: agg [100000*200] f32, then indeg [100000] f32 (80.4 MB)
    float* agg   = (float*)d_ws;
    float* indeg = agg + (size_t)NNODES * HDIM;

    const int zeroN4 = (NNODES * HDIM + NNODES) / 4;   // 5,025,000 float4
    zero_ws_kernel<<<2048, 256, 0, stream>>>((float4*)d_ws, zeroN4);

    edge_kernel<<<2048, 256, 0, stream>>>(dyn, wrd, weight, src, dst, ety, agg, indeg);

    finalize_kernel<<<(NUM_ENTS + 7) / 8, 256, 0, stream>>>(agg, indeg, out);
}